// DifferentiableEMA_1202590843039
// MI455X (gfx1250) — compile-verified
//
#include <hip/hip_runtime.h>
#include <stdint.h>

// ---------------------------------------------------------------------------
// EMA with NaN forward-fill as a decoupled-lookback scan over an augmented
// affine monoid. Bandwidth-bound: 128 MB moved once => ~5.5us at 23.3 TB/s.
// CDNA5 path: GLOBAL_LOAD_ASYNC_TO_LDS_B128 staging (ASYNCcnt) + wave32
// shuffle scans. WMMA intentionally NOT used: no matrix structure, op is
// memory-bound, tensor cores cannot reduce bytes moved.
// (ema_scan_kernel placed first in the TU so the disasm snippet shows the
//  async staging instructions.)
// ---------------------------------------------------------------------------

#define TILE      4096        // elements per workgroup (16 KB)
#define NTHREADS  256         // 8 wave32s
#define ITEMS     16          // elements per thread (blocked)
#define NWAVES    (NTHREADS / 32)

// LDS skew: +4 dwords every 64 floats => blocked 16-float reads go from
// 8-way to worst-case 2-way bank conflicts. Pads land on 16B chunk
// boundaries so 16B async writes never straddle a pad.
#define LPADI(i)   ((i) + (((i) >> 6) << 2))
#define LDS_FLOATS (TILE + ((TILE >> 6) << 2))   // 4352 floats = 17408 B

#define AS1 __attribute__((address_space(1)))
#define AS3 __attribute__((address_space(3)))

typedef int v4i __attribute__((ext_vector_type(4)));   // 128-bit payload type

#if __has_builtin(__builtin_amdgcn_global_load_async_to_lds_b128) && \
    __has_builtin(__builtin_amdgcn_global_store_async_from_lds_b128)
#define HAVE_ASYNC 1
#else
#define HAVE_ASYNC 0
#endif

#if HAVE_ASYNC
#if __has_builtin(__builtin_amdgcn_s_wait_asynccnt)
#define ASYNC_WAIT() __builtin_amdgcn_s_wait_asynccnt(0)
#else
#define ASYNC_WAIT() asm volatile("s_wait_asynccnt 0" ::: "memory")
#endif
#endif

// Monoid element:
//  (As,Bs): affine map v -> As*v + Bs assuming a valid sample precedes
//  Bns    : result of segment assuming NOTHING valid precedes (reset path)
//  hv     : segment contains at least one valid sample
struct SState { float As, Bs, Bns; int hv; };

// Lookback record: separate slots for aggregate and inclusive prefix so a
// reader can never confuse an upgraded payload (no single-slot race).
struct TileSlot { float As, Bs, Bns; int hv; int flag; int pad[3]; }; // 32 B
struct TileRec  { TileSlot agg; TileSlot pre; };                      // 64 B

__device__ __forceinline__ SState s_ident() {
  SState s; s.As = 1.0f; s.Bs = 0.0f; s.Bns = 0.0f; s.hv = 0; return s;
}

// p earlier, q later
__device__ __forceinline__ SState s_comb(SState p, SState q) {
  SState r;
  r.As  = p.As * q.As;
  r.Bs  = __builtin_fmaf(p.Bs, q.As, q.Bs);
  r.Bns = p.hv ? __builtin_fmaf(p.Bns, q.As, q.Bs) : q.Bns;
  r.hv  = p.hv | q.hv;
  return r;
}

__device__ __forceinline__ SState s_leaf(float x, float alpha, float oma) {
  SState s;
  const bool valid = (x == x);            // !isnan
  s.As  = valid ? oma : 1.0f;             // NaN: carry previous value
  s.Bs  = valid ? alpha * x : 0.0f;
  s.Bns = valid ? x : 0.0f;               // reset path: first valid => x
  s.hv  = valid ? 1 : 0;
  return s;
}

__device__ __forceinline__ SState s_shfl_up(SState v, int d) {
  SState r;
  r.As  = __shfl_up(v.As,  (unsigned)d, 32);
  r.Bs  = __shfl_up(v.Bs,  (unsigned)d, 32);
  r.Bns = __shfl_up(v.Bns, (unsigned)d, 32);
  r.hv  = __shfl_up(v.hv,  (unsigned)d, 32);
  return r;
}

__device__ __forceinline__ void slot_publish(TileSlot* s, SState v) {
  __hip_atomic_store(&s->As,  v.As,  __ATOMIC_RELAXED, __HIP_MEMORY_SCOPE_AGENT);
  __hip_atomic_store(&s->Bs,  v.Bs,  __ATOMIC_RELAXED, __HIP_MEMORY_SCOPE_AGENT);
  __hip_atomic_store(&s->Bns, v.Bns, __ATOMIC_RELAXED, __HIP_MEMORY_SCOPE_AGENT);
  __hip_atomic_store(&s->hv,  v.hv,  __ATOMIC_RELAXED, __HIP_MEMORY_SCOPE_AGENT);
  __hip_atomic_store(&s->flag, 1,    __ATOMIC_RELEASE, __HIP_MEMORY_SCOPE_AGENT);
}

__device__ __forceinline__ bool slot_try_read(TileSlot* s, SState* out) {
  if (__hip_atomic_load(&s->flag, __ATOMIC_ACQUIRE, __HIP_MEMORY_SCOPE_AGENT) == 0)
    return false;
  out->As  = __hip_atomic_load(&s->As,  __ATOMIC_RELAXED, __HIP_MEMORY_SCOPE_AGENT);
  out->Bs  = __hip_atomic_load(&s->Bs,  __ATOMIC_RELAXED, __HIP_MEMORY_SCOPE_AGENT);
  out->Bns = __hip_atomic_load(&s->Bns, __ATOMIC_RELAXED, __HIP_MEMORY_SCOPE_AGENT);
  out->hv  = __hip_atomic_load(&s->hv,  __ATOMIC_RELAXED, __HIP_MEMORY_SCOPE_AGENT);
  return true;
}

__global__ __launch_bounds__(NTHREADS)
void ema_scan_kernel(const float* __restrict__ x, const float* __restrict__ w_alpha,
                     float* __restrict__ out, int* counter, TileRec* recs) {
  __shared__ float  stage[LDS_FLOATS];
  __shared__ SState s_wagg[NWAVES];
  __shared__ SState s_wex[NWAVES];
  __shared__ SState s_tileex;
  __shared__ int    s_vtile;

  const int tid  = threadIdx.x;
  const int lane = tid & 31;
  const int wv   = tid >> 5;

  // Ticket: virtual tile id in launch order -> lookback forward progress.
  if (tid == 0)
    s_vtile = __hip_atomic_fetch_add(counter, 1, __ATOMIC_RELAXED, __HIP_MEMORY_SCOPE_AGENT);
  __syncthreads();
  const int tile = s_vtile;

  const float w     = w_alpha[0];
  const float alpha = 1.0f / (1.0f + __expf(-w));   // sigmoid
  const float oma   = 1.0f - alpha;

  const float* gsrc = x + (size_t)tile * TILE;

  // ---- Stage tile into LDS (coalesced 16B/lane). CDNA5 async path. ----
#if HAVE_ASYNC
  {
    const uint32_t lbase = (uint32_t)(uintptr_t)(&stage[0]); // low 32 = LDS offset
#pragma unroll
    for (int r = 0; r < 4; ++r) {
      const uint32_t c    = (uint32_t)(r * NTHREADS + tid);      // 16B chunk id
      const uint32_t loff = 16u * c + 16u * (c >> 4);            // skewed dest
      __builtin_amdgcn_global_load_async_to_lds_b128(
          (AS1 v4i*)(uintptr_t)(gsrc + 4u * c),
          (AS3 v4i*)(uint32_t)(lbase + loff), 0, 0);
    }
    ASYNC_WAIT();
  }
#else
#pragma unroll
  for (int r = 0; r < 4; ++r) {
    const int c = r * NTHREADS + tid;
    float4 v = *(const float4*)(gsrc + 4 * c);
    *(float4*)&stage[LPADI(4 * c)] = v;
  }
#endif
  __syncthreads();

  // ---- Pass 1: thread-serial aggregate over 16 blocked elements ----
  const int ebase = tid * ITEMS;
  SState ta = s_ident();
#pragma unroll
  for (int k = 0; k < ITEMS; ++k)
    ta = s_comb(ta, s_leaf(stage[LPADI(ebase + k)], alpha, oma));

  // Wave32 inclusive shuffle scan
  SState incl = ta;
#pragma unroll
  for (int d = 1; d < 32; d <<= 1) {
    SState o = s_shfl_up(incl, d);
    if (lane >= d) incl = s_comb(o, incl);
  }
  SState laneEx = s_shfl_up(incl, 1);
  if (lane == 0) laneEx = s_ident();
  if (lane == 31) s_wagg[wv] = incl;
  __syncthreads();

  // ---- Cross-wave scan + decoupled lookback (thread 0) ----
  if (tid == 0) {
    SState run = s_ident();
#pragma unroll
    for (int i = 0; i < NWAVES; ++i) { s_wex[i] = run; run = s_comb(run, s_wagg[i]); }

    TileRec* R = &recs[tile];
    slot_publish(&R->agg, run);                 // aggregate available ASAP

    SState ex;
    if (tile == 0) {
      ex = s_ident();
    } else {
      SState suffix = s_ident();
      int k = tile - 1;
      for (;;) {
        SState p;
        if (slot_try_read(&recs[k].pre, &p)) { ex = s_comb(p, suffix); break; }
        if (slot_try_read(&recs[k].agg, &p)) {
          suffix = s_comb(p, suffix);
          if (k == 0) { ex = suffix; break; }
          --k;
          continue;
        }
        __builtin_amdgcn_s_sleep(1);
      }
    }
    slot_publish(&R->pre, s_comb(ex, run));     // inclusive prefix
    s_tileex = ex;
  }
  __syncthreads();

  // ---- Pass 2: reapply with full exclusive prefix, stage results ----
  SState runst = s_comb(s_tileex, s_comb(s_wex[wv], laneEx));
  const float NANF = __int_as_float(0x7fc00000);
#pragma unroll
  for (int k = 0; k < ITEMS; ++k) {
    const int i = ebase + k;
    runst = s_comb(runst, s_leaf(stage[LPADI(i)], alpha, oma));
    stage[LPADI(i)] = runst.hv ? runst.Bns : NANF;   // NaN before first valid
  }
  __syncthreads();

  // ---- Drain LDS -> global (coalesced, async store path) ----
  float* gdst = out + (size_t)tile * TILE;
#if HAVE_ASYNC
  {
    const uint32_t lbase = (uint32_t)(uintptr_t)(&stage[0]);
#pragma unroll
    for (int r = 0; r < 4; ++r) {
      const uint32_t c    = (uint32_t)(r * NTHREADS + tid);
      const uint32_t loff = 16u * c + 16u * (c >> 4);
      __builtin_amdgcn_global_store_async_from_lds_b128(
          (AS1 v4i*)(uintptr_t)(gdst + 4u * c),
          (AS3 v4i*)(uint32_t)(lbase + loff), 0, 0);
    }
    ASYNC_WAIT();   // belt-and-braces; S_ENDPGM also waits idle
  }
#else
#pragma unroll
  for (int r = 0; r < 4; ++r) {
    const int c = r * NTHREADS + tid;
    *(float4*)(gdst + 4 * c) = *(const float4*)&stage[LPADI(4 * c)];
  }
#endif
}

__global__ void ema_init_kernel(int* counter, TileRec* recs, int ntiles) {
  int i = blockIdx.x * blockDim.x + threadIdx.x;
  if (i == 0)
    __hip_atomic_store(counter, 0, __ATOMIC_RELAXED, __HIP_MEMORY_SCOPE_AGENT);
  if (i < ntiles) {
    __hip_atomic_store(&recs[i].agg.flag, 0, __ATOMIC_RELAXED, __HIP_MEMORY_SCOPE_AGENT);
    __hip_atomic_store(&recs[i].pre.flag, 0, __ATOMIC_RELAXED, __HIP_MEMORY_SCOPE_AGENT);
  }
}

extern "C" void kernel_launch(void* const* d_in, const int* in_sizes, int n_in,
                              void* d_out, int out_size, void* d_ws, size_t ws_size,
                              hipStream_t stream) {
  (void)n_in; (void)out_size; (void)ws_size;
  const float* x = (const float*)d_in[0];
  const float* w = (const float*)d_in[1];
  float* out     = (float*)d_out;

  const int n      = in_sizes[0];       // 16,777,216 -> divisible by TILE
  const int ntiles = n / TILE;          // 4096

  int*     counter = (int*)d_ws;
  TileRec* recs    = (TileRec*)((char*)d_ws + 256);   // 4096*64B = 256 KB

  ema_init_kernel<<<(ntiles + 255) / 256, 256, 0, stream>>>(counter, recs, ntiles);
  ema_scan_kernel<<<ntiles, NTHREADS, 0, stream>>>(x, w, out, counter, recs);
}